// Conv2d_NN_Spatial_44976897523814
// MI455X (gfx1250) — compile-verified
//
#include <hip/hip_runtime.h>
#include <math.h>

// Problem constants (match reference)
#define BATCH 16
#define CIN   16
#define COUT  16
#define HH    160
#define WW    160
#define KK    9
#define HP    80
#define NP    (HP*HP)     // 6400 positions per batch
#define C1    72          // unshuffled channels
#define M_ANCH 256        // anchors
#define NCOLS 336         // 256 (inner products) + 72 (W0^T) + 8 pad
#define KSTEPS 18         // 72 / 4 (WMMA f32 K=4)

typedef __attribute__((ext_vector_type(2))) float    v2f;
typedef __attribute__((ext_vector_type(8))) float    v8f;
typedef __attribute__((ext_vector_type(4))) unsigned v4u;
typedef __attribute__((ext_vector_type(4))) int      v4i;
typedef __attribute__((ext_vector_type(8))) int      v8i;

// round(linspace(0,79,16)) — anchor coordinates on the 80x80 grid
__constant__ int c_XI[16] = {0,5,11,16,21,26,32,37,42,47,53,58,63,68,74,79};

// ---------------------------------------------------------------------------
// TDM: async-load a contiguous [rows x 72 f32] tile (row stride 72) from
// global memory into LDS via the Tensor Data Mover. D# per ISA 08 §8.
// Tensor is framed as exactly the tile, so no OOB cases arise.
// ---------------------------------------------------------------------------
#if __has_builtin(__builtin_amdgcn_tensor_load_to_lds)
#define HAVE_TDM 1
__device__ __forceinline__ void tdm_load_a_tile(void* lds_dst, const void* gsrc,
                                                unsigned rows) {
  unsigned long long ga = (unsigned long long)gsrc;
  unsigned lds_off = (unsigned)(unsigned long long)lds_dst; // addr[31:0] = LDS offset

  v4u g0;
  g0[0] = 1u;                                            // count=1, user D#
  g0[1] = lds_off;                                       // lds_addr (bytes)
  g0[2] = (unsigned)(ga & 0xFFFFFFFFu);                  // global_addr[31:0]
  g0[3] = (unsigned)((ga >> 32) & 0x01FFFFFFu)           // global_addr[56:32]
          | 0x80000000u;                                 // type=2 ("image")

  v8i g1;
  g1[0] = (int)(2u << 16);          // wg_mask=0 | data_size=2 (4 bytes)
  g1[1] = (int)(72u << 16);         // tensor_dim0 = 72       -> bits 79:48
  g1[2] = (int)(rows << 16);        // tensor_dim1 = rows     -> bits 111:80
  g1[3] = (int)(72u << 16);         // tile_dim0   = 72       -> bits 127:112
  g1[4] = (int)rows;                // tile_dim1 = rows, tile_dim2 = 0
  g1[5] = 72;                       // tensor_dim0_stride = 72 elements
  g1[6] = 0;                        // stride hi | tensor_dim1_stride lo
  g1[7] = 0;

  v4i gz = {0, 0, 0, 0};            // groups 2/3: unused for 2D tile
#if defined(__clang_major__) && (__clang_major__ >= 23)
  v8i gz8 = {0, 0, 0, 0, 0, 0, 0, 0};
  __builtin_amdgcn_tensor_load_to_lds(g0, g1, gz, gz, gz8, 0);
#else
  __builtin_amdgcn_tensor_load_to_lds(g0, g1, gz, gz, 0);
#endif
}
#else
#define HAVE_TDM 0
#endif

// ---------------------------------------------------------------------------
// K1: build x1T[b][n][c1] (position-major, channel-contiguous) = pixel-unshuffle
//     of (x ++ normalized coord channels); also per-position squared norm.
// ---------------------------------------------------------------------------
__global__ void k_prep(const float* __restrict__ x,
                       float* __restrict__ x1T, float* __restrict__ xsq) {
  int tid = blockIdx.x * blockDim.x + threadIdx.x;
  if (tid >= BATCH * NP) return;
  int b = tid / NP, n = tid % NP;
  int i = n / HP, j = n % HP;
  float s = 0.f;
  float* row = x1T + (size_t)tid * C1;
  #pragma unroll
  for (int c1 = 0; c1 < C1; ++c1) {
    int c = c1 >> 2, sy = (c1 >> 1) & 1, sx = c1 & 1;
    int h = 2*i + sy, w = 2*j + sx;
    float v;
    if (c < CIN) {
      v = x[(((size_t)b*CIN + c)*HH + h)*WW + w];
    } else {
      float fh = (float)h, fw = (float)w;
      float dn = fmaxf(sqrtf(fh*fh + fw*fw), 1e-12f);
      v = (c == CIN ? fh : fw) / dn;
    }
    row[c1] = v;
    s += v*v;
  }
  xsq[tid] = s;
}

// ---------------------------------------------------------------------------
// K2: build GEMM B operand, column-major-by-k: Bmat[b][col][k]
//     cols 0..255  = anchor columns xs (row copy from x1T)
//     cols 256..327= W0^T (conv_w[:,:,0])
//     cols 328..335= zero pad
//     Also ysq (anchor squared norms).
// ---------------------------------------------------------------------------
__global__ void k_bmat(const float* __restrict__ x1T, const float* __restrict__ xsq,
                       const float* __restrict__ conv_w,
                       float* __restrict__ Bmat, float* __restrict__ ysq) {
  int tid = blockIdx.x * blockDim.x + threadIdx.x;
  if (tid >= BATCH * NCOLS) return;
  int b = tid / NCOLS, col = tid % NCOLS;
  float* dst = Bmat + (size_t)tid * C1;
  if (col < M_ANCH) {
    int af = c_XI[col >> 4] * HP + c_XI[col & 15];
    const float* src = x1T + ((size_t)b*NP + af) * C1;
    #pragma unroll
    for (int c = 0; c < C1; ++c) dst[c] = src[c];
    ysq[b*M_ANCH + col] = xsq[b*NP + af];
  } else if (col < M_ANCH + C1) {
    int o = col - M_ANCH;
    #pragma unroll
    for (int c = 0; c < C1; ++c) dst[c] = conv_w[((size_t)o*C1 + c)*KK + 0];
  } else {
    #pragma unroll
    for (int c = 0; c < C1; ++c) dst[c] = 0.f;
  }
}

// ---------------------------------------------------------------------------
// K3a: pack W_k (k=1..8) transposed, k-contiguous: WkT[k-1][o(80 pad)][c]
// ---------------------------------------------------------------------------
__global__ void k_wkt(const float* __restrict__ conv_w, float* __restrict__ WkT) {
  int tid = blockIdx.x * blockDim.x + threadIdx.x;
  if (tid >= 8 * 80 * C1) return;
  int c = tid % C1;
  int o = (tid / C1) % 80;
  int kk = tid / (C1 * 80);
  WkT[tid] = (o < C1) ? conv_w[((size_t)o*C1 + c)*KK + (kk + 1)] : 0.f;
}

// ---------------------------------------------------------------------------
// K3b: Y[b][k][m][o] = sum_c W_{k+1}[o][c] * xs[b][c][m]   (WMMA f32 16x16x4)
//      A = anchor rows (Bmat rows 0..255), B = WkT columns.
// ---------------------------------------------------------------------------
__global__ void __launch_bounds__(32) k_ygemm(const float* __restrict__ Bmat,
                                              const float* __restrict__ WkT,
                                              float* __restrict__ Y) {
  int idx = blockIdx.x;
  int nt = idx % 5;  idx /= 5;    // o tile (80/16)
  int mt = idx % 16; idx /= 16;   // m tile (256/16)
  int kk = idx % 8;  int b = idx / 8;
  int lane = threadIdx.x;
  int half = lane >> 4, lm = lane & 15;
  int m0 = mt * 16, o0 = nt * 16;
  const float* A  = Bmat + (size_t)b * NCOLS * C1;   // rows 0..255 = anchors
  const float* Bp = WkT  + (size_t)kk * 80 * C1;
  v8f acc = {};
  for (int ks = 0; ks < KSTEPS; ++ks) {
    int k0 = 4*ks + 2*half;
    v2f a, bb;
    const float* ap = A  + (size_t)(m0 + lm) * C1 + k0;
    a.x = ap[0];  a.y = ap[1];
    const float* bp = Bp + (size_t)(o0 + lm) * C1 + k0;
    bb.x = bp[0]; bb.y = bp[1];
    acc = __builtin_amdgcn_wmma_f32_16x16x4_f32(false, a, false, bb,
                                                (short)0, acc, false, false);
  }
  int o = o0 + lm;
  if (o < C1) {
    #pragma unroll
    for (int r = 0; r < 8; ++r) {
      int m = m0 + r + 8*half;
      Y[(((size_t)b*8 + kk)*M_ANCH + m)*C1 + o] = acc[r];
    }
  }
}

// ---------------------------------------------------------------------------
// K4: mega-fused per 16-position tile (one wave32 per block):
//   TDM async A-tile load -> WMMA GEMM [16x72]x[72x336]
//   -> distances -> top-8 anchors -> out1d accumulation via Y lookups
//   -> pixel-shuffle + 16x18 pointwise -> final output.
// ---------------------------------------------------------------------------
__global__ void __launch_bounds__(32) k_main(
    const float* __restrict__ x1T, const float* __restrict__ xsq,
    const float* __restrict__ Bmat, const float* __restrict__ ysq,
    const float* __restrict__ Y,
    const float* __restrict__ conv_b, const float* __restrict__ pw_w,
    const float* __restrict__ pw_b, float* __restrict__ out) {
  __shared__ float As[16][C1];       // A tile: 16 positions x 72 channels
  __shared__ float Res[16][NCOLS];   // GEMM result tile

  int blk  = blockIdx.x;
  int b    = blk / (NP/16);
  int tile = blk % (NP/16);
  int n0   = tile * 16;
  int lane = threadIdx.x;

  const float* Bm = Bmat + (size_t)b * NCOLS * C1;
  __builtin_prefetch(Bm, 0, 0);      // warm the B panel (global_prefetch_b8)

#if HAVE_TDM
  // One TDM op per wave moves the whole 16x72 f32 tile into LDS.
  tdm_load_a_tile(&As[0][0], x1T + ((size_t)b*NP + n0) * C1, 16u);
  __builtin_amdgcn_s_wait_tensorcnt(0);
#else
  for (int t = lane; t < 16*C1; t += 32)
    As[t / C1][t % C1] = x1T[((size_t)b*NP + n0 + t/C1)*C1 + (t % C1)];
#endif
  __syncthreads();

  int half = lane >> 4, lm = lane & 15;

  // Hoist A fragments for all 18 K-steps into registers
  v2f afr[KSTEPS];
  #pragma unroll
  for (int ks = 0; ks < KSTEPS; ++ks) {
    afr[ks].x = As[lm][4*ks + 2*half];
    afr[ks].y = As[lm][4*ks + 2*half + 1];
  }

  for (int ct = 0; ct < NCOLS/16; ++ct) {            // 21 column tiles
    v8f acc = {};
    const float* bp0 = Bm + (size_t)(ct*16 + lm) * C1 + 2*half;
    #pragma unroll
    for (int ks = 0; ks < KSTEPS; ++ks) {
      v2f bb;
      bb.x = bp0[4*ks]; bb.y = bp0[4*ks + 1];
      acc = __builtin_amdgcn_wmma_f32_16x16x4_f32(false, afr[ks], false, bb,
                                                  (short)0, acc, false, false);
    }
    #pragma unroll
    for (int r = 0; r < 8; ++r)
      Res[r + 8*half][ct*16 + lm] = acc[r];
  }
  __syncthreads();

  if (lane < 16) {
    int pos = n0 + lane;
    float xq = xsq[b*NP + pos];
    // top-8 smallest squared distances (sorted ascending, stable on ties)
    float bd[8]; int bi[8];
    #pragma unroll
    for (int s = 0; s < 8; ++s) { bd[s] = 3.0e38f; bi[s] = 0; }
    const float* yq = ysq + b * M_ANCH;
    for (int m = 0; m < M_ANCH; ++m) {
      float d2 = fmaxf(xq + yq[m] - 2.f * Res[lane][m], 0.f);
      int af = c_XI[m >> 4] * HP + c_XI[m & 15];
      if (af == pos) d2 = 3.4e38f;                   // self-match exclusion
      float v = d2; int vi = m;
      #pragma unroll
      for (int s = 0; s < 8; ++s) {
        if (v < bd[s]) {
          float tf = bd[s]; int ti = bi[s];
          bd[s] = v; bi[s] = vi; v = tf; vi = ti;
        }
      }
    }
    // out1d column = W0-term + bias + sum of 8 precomputed anchor responses
    float acc72[C1];
    #pragma unroll
    for (int o = 0; o < C1; ++o) acc72[o] = Res[lane][M_ANCH + o] + conv_b[o];
    #pragma unroll
    for (int k = 0; k < 8; ++k) {
      const float* yc = Y + (((size_t)b*8 + k)*M_ANCH + bi[k]) * C1;
      #pragma unroll
      for (int o = 0; o < C1; ++o) acc72[o] += yc[o];
    }
    // pixel-shuffle + pointwise (16 out ch x 18 in ch) fused store
    int i = pos / HP, j = pos % HP;
    #pragma unroll
    for (int sub = 0; sub < 4; ++sub) {
      int h = 2*i + (sub >> 1), w = 2*j + (sub & 1);
      #pragma unroll
      for (int o2 = 0; o2 < COUT; ++o2) {
        float s = pw_b[o2];
        #pragma unroll
        for (int c2 = 0; c2 < 18; ++c2)
          s += pw_w[o2*18 + c2] * acc72[c2*4 + sub];
        out[(((size_t)b*COUT + o2)*HH + h)*WW + w] = s;
      }
    }
  }
}

// ---------------------------------------------------------------------------
extern "C" void kernel_launch(void* const* d_in, const int* in_sizes, int n_in,
                              void* d_out, int out_size, void* d_ws, size_t ws_size,
                              hipStream_t stream) {
  (void)in_sizes; (void)n_in; (void)out_size; (void)ws_size;
  const float* x      = (const float*)d_in[0];
  const float* conv_w = (const float*)d_in[1];
  const float* conv_b = (const float*)d_in[2];
  const float* pw_w   = (const float*)d_in[3];
  const float* pw_b   = (const float*)d_in[4];
  float* out = (float*)d_out;

  // Workspace carve-up (floats): total ~10.3M floats (~41 MB)
  float* ws   = (float*)d_ws;
  size_t off  = 0;
  float* x1T  = ws + off; off += (size_t)BATCH * NP * C1;
  float* xsq  = ws + off; off += (size_t)BATCH * NP;
  float* Bmat = ws + off; off += (size_t)BATCH * NCOLS * C1;
  float* ysq  = ws + off; off += (size_t)BATCH * M_ANCH;
  float* WkT  = ws + off; off += (size_t)8 * 80 * C1;
  float* Ybuf = ws + off; off += (size_t)BATCH * 8 * M_ANCH * C1;

  k_prep <<<(BATCH*NP    + 255)/256, 256, 0, stream>>>(x, x1T, xsq);
  k_bmat <<<(BATCH*NCOLS + 255)/256, 256, 0, stream>>>(x1T, xsq, conv_w, Bmat, ysq);
  k_wkt  <<<(8*80*C1     + 255)/256, 256, 0, stream>>>(conv_w, WkT);
  k_ygemm<<<BATCH*8*16*5, 32, 0, stream>>>(Bmat, WkT, Ybuf);
  k_main <<<BATCH*(NP/16), 32, 0, stream>>>(x1T, xsq, Bmat, ysq, Ybuf,
                                            conv_b, pw_w, pw_b, out);
}